// LaBGATr_53549652246789
// MI455X (gfx1250) — compile-verified
//
#include <hip/hip_runtime.h>

// ---------------------------------------------------------------------------
// LaBGATr interp + equivariant MLP, fused, fp32 WMMA (V_WMMA_F32_16X16X4_F32)
// Round 3: wave-local LDS fences (s_wait_dscnt) instead of workgroup barriers;
//          all LDS scratch is per-wave, so the 4 waves run fully decoupled.
// ---------------------------------------------------------------------------

typedef __attribute__((ext_vector_type(2))) float v2f;
typedef __attribute__((ext_vector_type(4))) float v4f;
typedef __attribute__((ext_vector_type(8))) float v8f;

#define N_SRC   16384
#define N_TGT   65536
#define KNN     8
#define C_MV    16
#define C_SKIP  2
#define C_CAT   18          // C_MV + C_SKIP
#define C_PAD   20          // padded K for 5 x (K=4) WMMA steps
#define S_SC    32
#define HID     64
#define MV      16

#define TGT_PER_WAVE 16
#define WAVES        4
#define TGT_PER_WG   (TGT_PER_WAVE * WAVES)

#define HPAD 66             // mv_h / scA LDS row stride (bank-conflict free)

// Wave-local LDS fence: same-wave LDS ops are processed in order by the LDS
// unit (ISA 7.3); waiting DScnt==0 + a compiler memory barrier is sufficient
// for cross-lane dataflow through per-wave LDS buffers. No s_barrier needed.
#define WAVE_LDS_FENCE() asm volatile("s_wait_dscnt 0x0" ::: "memory")

static __device__ __forceinline__ v8f wmma4(v2f a, v2f b, v8f c) {
    // D = A(16x4 f32) * B(4x16 f32) + C(16x16 f32)
    return __builtin_amdgcn_wmma_f32_16x16x4_f32(
        /*neg_a=*/false, a, /*neg_b=*/false, b,
        /*c_mod=*/(short)0, c, /*reuse_a=*/false, /*reuse_b=*/false);
}

__global__ __launch_bounds__(128, 1) void labgatr_fused(
    const float* __restrict__ mv_src,  const float* __restrict__ mv_skip,
    const float* __restrict__ sc_src,  const float* __restrict__ sc_skip,
    const float* __restrict__ pos_src, const float* __restrict__ pos_tgt,
    const int*   __restrict__ isrc,
    const float* __restrict__ W1mv, const float* __restrict__ W2mv,
    const float* __restrict__ W1s,  const float* __restrict__ b1s,
    const float* __restrict__ W2s,  const float* __restrict__ b2s,
    float* __restrict__ out_mv, float* __restrict__ out_sc)
{
    // per-wave LDS scratch only; all weights live in regs / L2
    __shared__ float sMvT[WAVES][MV * C_PAD];            // A staging (blade x padded-ch) + out transpose
    __shared__ float sMvH[WAVES][MV * HPAD];             // mv_h (blade x hidden)
    __shared__ float sScA[WAVES][TGT_PER_WAVE * HPAD];   // scalar activations (target x 2S / hid)

    const int tid  = threadIdx.x;
    const int lane = tid & 31;
    // force wave id into an SGPR so per-target edge metadata becomes SMEM loads
    const int wave = __builtin_amdgcn_readfirstlane(tid >> 5);
    const int colL = lane & 15;            // N column inside a 16-wide tile
    const int kOff = (lane >> 4) * 2;      // K sub-offset of f32 16x16x4 A/B frags
    const int mRow = lane & 15;            // A-matrix row for this lane
    const int rowD = (lane >= 16) ? 8 : 0; // D-layout row offset

    float* mvT = sMvT[wave];
    float* mvh = sMvH[wave];
    float* scA = sScA[wave];

    // ------- hoist weight B-fragments into registers (read once, L2-hot) -------
    v2f b1f[5][4];                         // W1_mv: (18->20) x 64, 4 col tiles
#pragma unroll
    for (int kb = 0; kb < 5; ++kb) {
        const int r0 = kb * 4 + kOff, r1 = r0 + 1;
#pragma unroll
        for (int j = 0; j < 4; ++j) {
            b1f[kb][j].x = (r0 < C_CAT) ? W1mv[r0 * HID + j * 16 + colL] : 0.0f;
            b1f[kb][j].y = (r1 < C_CAT) ? W1mv[r1 * HID + j * 16 + colL] : 0.0f;
        }
    }
    v2f b2f[16];                           // W2_mv: 64 x 16
#pragma unroll
    for (int kk = 0; kk < 16; ++kk) {
        b2f[kk].x = W2mv[(kk * 4 + kOff)     * C_MV + colL];
        b2f[kk].y = W2mv[(kk * 4 + kOff + 1) * C_MV + colL];
    }

    const int n0 = blockIdx.x * TGT_PER_WG + wave * TGT_PER_WAVE;   // wave-uniform (SGPR)

    for (int t = 0; t < TGT_PER_WAVE; ++t) {
        const int n = n0 + t;                          // uniform -> s_load for edge data
        WAVE_LDS_FENCE();                              // mvT reuse fence (prev iter reads done)

        if (t + 1 < TGT_PER_WAVE)                      // hint next target's edge list
            __builtin_prefetch(&isrc[(n + 1) * KNN], 0, 3);

        // ---------------- IDW interpolation (K=8 contiguous edges) ----------------
        const float px = pos_tgt[n * 3 + 0];
        const float py = pos_tgt[n * 3 + 1];
        const float pz = pos_tgt[n * 3 + 2];

        v4f acc0 = {}, acc1 = {};                      // 8 mv elements / lane, b128 gathers
        float sacc = 0.0f, denom = 0.0f;

#pragma unroll
        for (int k = 0; k < KNN; ++k) {
            const int   s  = isrc[n * KNN + k];        // uniform -> SMEM
            const float dx = pos_src[s * 3 + 0] - px;  // uniform base -> SMEM
            const float dy = pos_src[s * 3 + 1] - py;
            const float dz = pos_src[s * 3 + 2] - pz;
            const float d2 = fmaxf(dx * dx + dy * dy + dz * dz, 1e-16f);
            const float w  = 1.0f / d2;
            denom += w;
            const v4f* row4 = (const v4f*)(mv_src + (size_t)s * (C_MV * MV));
            const v4f  r0 = row4[lane];                // elements 4L..4L+3   (global_load_b128)
            const v4f  r1 = row4[32 + lane];           // elements 128+4L..   (global_load_b128)
#pragma unroll
            for (int j = 0; j < 4; ++j) {
                acc0[j] = fmaf(w, r0[j], acc0[j]);
                acc1[j] = fmaf(w, r1[j], acc1[j]);
            }
            sacc = fmaf(w, sc_src[s * S_SC + lane], sacc);
        }
        const float invd = 1.0f / denom;

        // store A-matrix transposed: mvT[m][c] = mv_cat[c][m]
#pragma unroll
        for (int j = 0; j < 4; ++j) {
            const int e0 = lane * 4 + j;               // first 128 elements
            const int e1 = 128 + e0;                   // second 128 elements
            mvT[(e0 & 15) * C_PAD + (e0 >> 4)] = acc0[j] * invd;
            mvT[(e1 & 15) * C_PAD + (e1 >> 4)] = acc1[j] * invd;
        }
        {   // skip channels 16..17 + zero pad 18..19
            const float v = mv_skip[(size_t)n * (C_SKIP * MV) + lane];
            mvT[(lane & 15) * C_PAD + 16 + (lane >> 4)] = v;
            if (lane < 16) {
                mvT[lane * C_PAD + 18] = 0.0f;
                mvT[lane * C_PAD + 19] = 0.0f;
            }
        }
        // scalar concat row for this target
        scA[t * HPAD + lane]        = sacc * invd;
        scA[t * HPAD + S_SC + lane] = sc_skip[(size_t)n * S_SC + lane];
        WAVE_LDS_FENCE();                              // mvT writes visible to all lanes

        // ---------------- stage 1: (16 x 18pad20) @ (18 x 64) ----------------
        v8f h[4] = {};
#pragma unroll
        for (int kb = 0; kb < 5; ++kb) {
            v2f a;
            a.x = mvT[mRow * C_PAD + kb * 4 + kOff];
            a.y = mvT[mRow * C_PAD + kb * 4 + kOff + 1];
            h[0] = wmma4(a, b1f[kb][0], h[0]);
            h[1] = wmma4(a, b1f[kb][1], h[1]);
            h[2] = wmma4(a, b1f[kb][2], h[2]);
            h[3] = wmma4(a, b1f[kb][3], h[3]);
        }
        // D layout -> LDS as mv_h[blade][hid]
#pragma unroll
        for (int j = 0; j < 4; ++j)
#pragma unroll
            for (int r = 0; r < 8; ++r)
                mvh[(r + rowD) * HPAD + j * 16 + colL] = h[j][r];
        WAVE_LDS_FENCE();                              // mv_h visible

        // ---------------- stage 2: (16 x 64) @ (64 x 16) ----------------
        v8f o = {};
#pragma unroll
        for (int kk = 0; kk < 16; ++kk) {
            v2f a;
            a.x = mvh[mRow * HPAD + kk * 4 + kOff];
            a.y = mvh[mRow * HPAD + kk * 4 + kOff + 1];
            o = wmma4(a, b2f[kk], o);
        }
        // transpose-stage: out layout is mv_o[n][c][m]; D gives (m, c)
#pragma unroll
        for (int r = 0; r < 8; ++r)
            mvT[colL * 17 + (r + rowD)] = o[r];        // stride-17: conflict free
        WAVE_LDS_FENCE();                              // staging visible

        float* dst = out_mv + (size_t)n * (C_MV * MV);
#pragma unroll
        for (int it = 0; it < 8; ++it) {
            const int e = it * 32 + lane;
            dst[e] = mvT[(e >> 4) * 17 + (e & 15)];    // coalesced 128B stores
        }
    }

    // ================= scalar MLP: 16 targets of this wave as one GEMM =================
    WAVE_LDS_FENCE();
    // GEMM1: scA(16 x 64) @ W1_s(64 x 64)
    v8f g[4] = {};
#pragma unroll
    for (int kk = 0; kk < 16; ++kk) {
        const int r0 = kk * 4 + kOff, r1 = r0 + 1;
        v2f a;
        a.x = scA[mRow * HPAD + r0];
        a.y = scA[mRow * HPAD + r1];
#pragma unroll
        for (int j = 0; j < 4; ++j) {
            v2f b;
            b.x = W1s[r0 * HID + j * 16 + colL];
            b.y = W1s[r1 * HID + j * 16 + colL];
            g[j] = wmma4(a, b, g[j]);
        }
    }
    WAVE_LDS_FENCE();                                   // before overwriting scA with sc_h

    // bias + tanh-GELU on D fragments, restage as sc_h
#pragma unroll
    for (int j = 0; j < 4; ++j) {
        const float bias = b1s[j * 16 + colL];
#pragma unroll
        for (int r = 0; r < 8; ++r) {
            const float x  = g[j][r] + bias;
            const float u  = 0.7978845608028654f * (x + 0.044715f * x * x * x);
            scA[(r + rowD) * HPAD + j * 16 + colL] = 0.5f * x * (1.0f + tanhf(u));
        }
    }
    WAVE_LDS_FENCE();                                   // sc_h visible

    // GEMM2: sc_h(16 x 64) @ W2_s(64 x 32)
    v8f q0 = {}, q1 = {};
#pragma unroll
    for (int kk = 0; kk < 16; ++kk) {
        const int r0 = kk * 4 + kOff, r1 = r0 + 1;
        v2f a;
        a.x = scA[mRow * HPAD + r0];
        a.y = scA[mRow * HPAD + r1];
        v2f b0, b1v;
        b0.x  = W2s[r0 * S_SC + colL];       b0.y  = W2s[r1 * S_SC + colL];
        b1v.x = W2s[r0 * S_SC + 16 + colL];  b1v.y = W2s[r1 * S_SC + 16 + colL];
        q0 = wmma4(a, b0, q0);
        q1 = wmma4(a, b1v, q1);
    }
    const float bz0 = b2s[colL], bz1 = b2s[16 + colL];
#pragma unroll
    for (int r = 0; r < 8; ++r) {
        const int tt = r + rowD;
        out_sc[(size_t)(n0 + tt) * S_SC + colL]      = q0[r] + bz0;
        out_sc[(size_t)(n0 + tt) * S_SC + 16 + colL] = q1[r] + bz1;
    }
}

extern "C" void kernel_launch(void* const* d_in, const int* in_sizes, int n_in,
                              void* d_out, int out_size, void* d_ws, size_t ws_size,
                              hipStream_t stream) {
    const float* mv_src  = (const float*)d_in[0];
    const float* mv_skip = (const float*)d_in[1];
    const float* sc_src  = (const float*)d_in[2];
    const float* sc_skip = (const float*)d_in[3];
    const float* pos_src = (const float*)d_in[4];
    const float* pos_tgt = (const float*)d_in[5];
    const int*   isrc    = (const int*)d_in[6];
    // d_in[7] = interp_target: structurally repeat(arange(N_TGT), K) -> implicit
    const float* W1mv = (const float*)d_in[8];
    const float* W2mv = (const float*)d_in[9];
    const float* W1s  = (const float*)d_in[10];
    const float* b1s  = (const float*)d_in[11];
    const float* W2s  = (const float*)d_in[12];
    const float* b2s  = (const float*)d_in[13];

    float* out_mv = (float*)d_out;                       // [N_TGT, 16, 16]
    float* out_sc = out_mv + (size_t)N_TGT * C_MV * MV;  // [N_TGT, 32]

    dim3 grid(N_TGT / TGT_PER_WG);                       // 1024 blocks
    dim3 block(128);                                     // 4 wave32
    hipLaunchKernelGGL(labgatr_fused, grid, block, 0, stream,
                       mv_src, mv_skip, sc_src, sc_skip, pos_src, pos_tgt, isrc,
                       W1mv, W2mv, W1s, b1s, W2s, b2s, out_mv, out_sc);
}